// Net_27522150433131
// MI455X (gfx1250) — compile-verified
//
#include <hip/hip_runtime.h>
#include <hip/hip_bf16.h>

typedef __attribute__((ext_vector_type(2))) float v2f;
typedef __attribute__((ext_vector_type(8))) float v8f;

// ---------------------------------------------------------------------------
// Kernel 1: xw = x @ W1   (N x 128) * (128 x 64) in exact fp32 via
// V_WMMA_F32_16X16X4_F32. One wave -> 16 rows x 64 cols (4 accumulators).
// ---------------------------------------------------------------------------
__global__ __launch_bounds__(256) void gcn_gemm_xw(
    const float* __restrict__ x, const float* __restrict__ W1,
    float* __restrict__ xw, int nTiles, int N) {
  const int wave = threadIdx.x >> 5;   // wave32
  const int lane = threadIdx.x & 31;
  const int tile = blockIdx.x * 8 + wave;
  if (tile >= nTiles) return;

  const int row0 = tile * 16;
  const int m    = lane & 15;      // M index (A) / N index (B,C,D)
  const int half = lane >> 4;      // lane group 0/1
  const int kb   = half * 2;       // K sub-offset within group-of-4

  // clamp for (unused here) ragged tails; N is a multiple of 16 in practice
  const int arow = (row0 + m < N) ? (row0 + m) : (N - 1);
  const float* __restrict__ xrow = x + (size_t)arow * 128;

  v8f acc0 = {}, acc1 = {}, acc2 = {}, acc3 = {};

  for (int k0 = 0; k0 < 128; k0 += 4) {
    // A: 16x4 fp32 tile. lanes 0-15: K=k0,k0+1 ; lanes 16-31: K=k0+2,k0+3
    v2f a;
    a.x = xrow[k0 + kb];
    a.y = xrow[k0 + kb + 1];
    // B: 4x16 fp32 tiles, row-striped across lanes (N = lane&15)
    const float* __restrict__ w0 = W1 + (size_t)(k0 + kb) * 64;
    const float* __restrict__ w1 = W1 + (size_t)(k0 + kb + 1) * 64;
    v2f b0; b0.x = w0[ 0 + m]; b0.y = w1[ 0 + m];
    v2f b1; b1.x = w0[16 + m]; b1.y = w1[16 + m];
    v2f b2; b2.x = w0[32 + m]; b2.y = w1[32 + m];
    v2f b3; b3.x = w0[48 + m]; b3.y = w1[48 + m];

    acc0 = __builtin_amdgcn_wmma_f32_16x16x4_f32(false, a, false, b0, (short)0, acc0, false, false);
    acc1 = __builtin_amdgcn_wmma_f32_16x16x4_f32(false, a, false, b1, (short)0, acc1, false, false);
    acc2 = __builtin_amdgcn_wmma_f32_16x16x4_f32(false, a, false, b2, (short)0, acc2, false, false);
    acc3 = __builtin_amdgcn_wmma_f32_16x16x4_f32(false, a, false, b3, (short)0, acc3, false, false);
  }

  // C/D layout: VGPR r holds row (r + 8*half), col = lane&15 within 16-col tile
  #pragma unroll
  for (int r = 0; r < 8; ++r) {
    const int mm = row0 + r + 8 * half;
    if (mm < N) {
      float* __restrict__ o = xw + (size_t)mm * 64;
      o[ 0 + m] = acc0[r];
      o[16 + m] = acc1[r];
      o[32 + m] = acc2[r];
      o[48 + m] = acc3[r];
    }
  }
}

// ---------------------------------------------------------------------------
// Kernel 2: SpMM scatter. One wave per edge, 2 channels per lane.
// agg[dst] += xw[src] * w   via hardware global_atomic_add_f32.
// ---------------------------------------------------------------------------
__global__ __launch_bounds__(256) void gcn_edge_scatter(
    const int* __restrict__ esrc, const int* __restrict__ edst,
    const float* __restrict__ ew, const float* __restrict__ xw,
    float* __restrict__ agg, int E) {
  const long long gid = (long long)blockIdx.x * blockDim.x + threadIdx.x;
  const int e = (int)(gid >> 5);
  if (e >= E) return;
  const int lane = (int)(gid & 31);

  const int   s = esrc[e];
  const int   d = edst[e];
  const float w = ew[e];

  const float2 v = *((const float2*)(xw + (size_t)s * 64) + lane);
  float* __restrict__ o = agg + (size_t)d * 64 + lane * 2;
  unsafeAtomicAdd(o + 0, v.x * w);
  unsafeAtomicAdd(o + 1, v.y * w);
}

// ---------------------------------------------------------------------------
// Kernel 3: h = relu(agg + b1); segment-sum into sums/cnts.
// seg_ids is sorted -> register accumulation, flush on segment change.
// blockDim = 64 (one thread per channel), 256 nodes per block.
// ---------------------------------------------------------------------------
__global__ __launch_bounds__(64) void gcn_pool(
    const float* __restrict__ agg, const float* __restrict__ b1,
    const int* __restrict__ seg, float* __restrict__ sums,
    float* __restrict__ cnts, int N) {
  const int c  = threadIdx.x;       // channel
  const int n0 = blockIdx.x * 256;
  const int n1 = (n0 + 256 < N) ? (n0 + 256) : N;
  const float bias = b1[c];

  float acc = 0.f, cnt = 0.f;
  int cur = -1;
  for (int n = n0; n < n1; ++n) {
    const int g = seg[n];
    if (g != cur) {
      if (cur >= 0) {
        unsafeAtomicAdd(&sums[(size_t)cur * 64 + c], acc);
        if (c == 0) unsafeAtomicAdd(&cnts[cur], cnt);
      }
      acc = 0.f; cnt = 0.f; cur = g;
    }
    const float v = agg[(size_t)n * 64 + c] + bias;
    acc += fmaxf(v, 0.f);
    cnt += 1.f;
  }
  if (cur >= 0) {
    unsafeAtomicAdd(&sums[(size_t)cur * 64 + c], acc);
    if (c == 0) unsafeAtomicAdd(&cnts[cur], cnt);
  }
}

// ---------------------------------------------------------------------------
// Kernel 4: pooled = sums / max(cnt,1); logits = pooled @ W2 + b2; softmax.
// One thread per graph.
// ---------------------------------------------------------------------------
__global__ __launch_bounds__(64) void gcn_head(
    const float* __restrict__ sums, const float* __restrict__ cnts,
    const float* __restrict__ W2, const float* __restrict__ b2,
    float* __restrict__ out, int G) {
  const int g = threadIdx.x;
  if (g >= G) return;
  const float inv = 1.f / fmaxf(cnts[g], 1.f);

  float l0 = b2[0], l1 = b2[1], l2 = b2[2], l3 = b2[3];
  const float* __restrict__ s = sums + (size_t)g * 64;
  #pragma unroll 4
  for (int c = 0; c < 64; ++c) {
    const float p = s[c] * inv;
    l0 += p * W2[c * 4 + 0];
    l1 += p * W2[c * 4 + 1];
    l2 += p * W2[c * 4 + 2];
    l3 += p * W2[c * 4 + 3];
  }
  float mx = fmaxf(fmaxf(l0, l1), fmaxf(l2, l3));
  const float e0 = __expf(l0 - mx), e1 = __expf(l1 - mx);
  const float e2 = __expf(l2 - mx), e3 = __expf(l3 - mx);
  const float invs = 1.f / (e0 + e1 + e2 + e3);
  out[g * 4 + 0] = e0 * invs;
  out[g * 4 + 1] = e1 * invs;
  out[g * 4 + 2] = e2 * invs;
  out[g * 4 + 3] = e3 * invs;
}

// ---------------------------------------------------------------------------
extern "C" void kernel_launch(void* const* d_in, const int* in_sizes, int n_in,
                              void* d_out, int out_size, void* d_ws, size_t ws_size,
                              hipStream_t stream) {
  const float* x    = (const float*)d_in[0];
  const int*   esrc = (const int*)  d_in[1];
  const int*   edst = (const int*)  d_in[2];
  const float* ew   = (const float*)d_in[3];
  const int*   seg  = (const int*)  d_in[4];
  const float* W1   = (const float*)d_in[5];
  const float* b1   = (const float*)d_in[6];
  const float* W2   = (const float*)d_in[7];
  const float* b2   = (const float*)d_in[8];

  const int N = in_sizes[4];          // 100000 nodes
  const int E = in_sizes[1];          // 3.2M edges
  const int G = out_size / 4;         // 64 graphs
  float* out = (float*)d_out;

  float* xw   = (float*)d_ws;                  // N*64
  float* agg  = xw   + (size_t)N * 64;         // N*64
  float* sums = agg  + (size_t)N * 64;         // G*64
  float* cnts = sums + (size_t)G * 64;         // G

  // zero agg + sums + cnts (graph-capture safe)
  hipMemsetAsync(agg, 0, ((size_t)N * 64 + (size_t)G * 64 + G) * sizeof(float),
                 stream);

  const int nTiles = (N + 15) / 16;
  gcn_gemm_xw<<<(nTiles + 7) / 8, 256, 0, stream>>>(x, W1, xw, nTiles, N);

  const long long tE = (long long)E * 32;
  gcn_edge_scatter<<<(unsigned)((tE + 255) / 256), 256, 0, stream>>>(
      esrc, edst, ew, xw, agg, E);

  gcn_pool<<<(N + 255) / 256, 64, 0, stream>>>(agg, b1, seg, sums, cnts, N);

  gcn_head<<<1, 64, 0, stream>>>(sums, cnts, W2, b2, out, G);
}